// CTC_Attention_Loss_19035295056155
// MI455X (gfx1250) — compile-verified
//
#include <hip/hip_runtime.h>

// ---------------------------------------------------------------------------
// CTC + attention label-smoothing loss for MI455X (gfx1250, wave32).
//  Kernel A: label-smoothing loss; row sums computed as GEMV-vs-ones via
//            V_WMMA_F32_16X16X4_F32 (exact f32). A-tiles are staged into LDS
//            with GLOBAL_LOAD_ASYNC_TO_LDS_B128 (ASYNCcnt), double-buffered,
//            no barriers in the steady-state loop (wave-private tiles).
//  Kernel B: CTC alpha recursion, one workgroup per batch, alpha in LDS.
//  Kernel C: deterministic finalize.
// ---------------------------------------------------------------------------

typedef float v2f __attribute__((ext_vector_type(2)));
typedef float v8f __attribute__((ext_vector_type(8)));

#define VOCAB   5000
#define NEGV    (-1e30f)
#define ALPHA_W 0.2f
#define SMOOTH  0.1f

__device__ __forceinline__ float lae(float a, float b) {
    // matches jnp.logaddexp for finite inputs incl. the -1e30 sentinel
    float mx = fmaxf(a, b);
    float mn = fminf(a, b);
    return mx + log1pf(__expf(mn - mx));
}

// ---------------------------------------------------------------------------
// Kernel A: 32 blocks x 128 threads (4 waves). Each wave owns a 16-row tile.
// Chunk = 16 rows x 40 cols (125 chunks * 40 = 5000 exactly, no tail).
// Staging: per lane one async b128 copy x5 -> LDS (stride 68 dw: bank-safe).
// Compute: 10x v_wmma_f32_16x16x4_f32 with B = ones.
// ---------------------------------------------------------------------------
__global__ __launch_bounds__(128) void att_ls_kernel(const float* __restrict__ att,
                                                     const int*   __restrict__ targets,
                                                     float*       __restrict__ ws) {
    // [wave][buffer][16 rows * 68 dw]  (68 dw = 272 B, multiple of 16)
    __shared__ __align__(16) float tile[4][2][16 * 68];
    __shared__ float rowsum[64];
    __shared__ float perrow[64];

    const int tid     = threadIdx.x;
    const int w       = tid >> 5;
    const int lane    = tid & 31;
    const int rowbase = blockIdx.x * 64 + w * 16;

    const int lrow = lane & 15;          // A-matrix M for this lane
    const int koff = (lane >> 4) << 1;   // lanes 16-31 carry K=2,3

    v8f acc   = {0.f, 0.f, 0.f, 0.f, 0.f, 0.f, 0.f, 0.f};
    v2f bones = {1.0f, 1.0f};            // B = ones(4x16): every B VGPR is 1.0

    // Per-lane staging assignment: 160 float4 per chunk / 32 lanes = 5 each.
    // Precompute this lane's 5 (row, col4) pairs -> LDS byte offsets.
    unsigned lds_off[5];
    int      grow[5], gcol[5];
    #pragma unroll
    for (int j = 0; j < 5; ++j) {
        int f4 = j * 32 + lane;          // 0..159
        int r  = f4 / 10;
        int c4 = f4 - r * 10;
        grow[j] = r;
        gcol[j] = c4 * 4;
        // low 32 bits of a generic shared pointer == wave-relative LDS offset
        lds_off[j] = (unsigned)(uintptr_t)&tile[w][0][r * 68 + c4 * 4];
    }
    const unsigned buf_stride = (unsigned)(16 * 68 * sizeof(float)); // 4352 B

    // Issue chunk 0 into buffer 0.
    #pragma unroll
    for (int j = 0; j < 5; ++j) {
        unsigned long long ga = (unsigned long long)(uintptr_t)
            (att + (size_t)(rowbase + grow[j]) * VOCAB + gcol[j]);
        asm volatile("global_load_async_to_lds_b128 %0, %1, off"
                     :: "v"(lds_off[j]), "v"(ga) : "memory");
    }

    for (int chunk = 0; chunk < 125; ++chunk) {
        const unsigned cbuf = (unsigned)(chunk & 1) * buf_stride;
        if (chunk + 1 < 125) {
            const int k0n = (chunk + 1) * 40;
            const unsigned nbuf = (unsigned)((chunk + 1) & 1) * buf_stride;
            #pragma unroll
            for (int j = 0; j < 5; ++j) {
                unsigned long long ga = (unsigned long long)(uintptr_t)
                    (att + (size_t)(rowbase + grow[j]) * VOCAB + k0n + gcol[j]);
                asm volatile("global_load_async_to_lds_b128 %0, %1, off"
                             :: "v"(lds_off[j] + nbuf), "v"(ga) : "memory");
            }
            // in-order async completion: <=5 outstanding => chunk's 5 are done
            asm volatile("s_wait_asynccnt 0x5" ::: "memory");
        } else {
            asm volatile("s_wait_asynccnt 0x0" ::: "memory");
        }

        const float* tb = &tile[w][0][0] + (cbuf >> 2);
        #pragma unroll
        for (int kk = 0; kk < 40; kk += 4) {
            v2f a = *reinterpret_cast<const v2f*>(&tb[lrow * 68 + kk + koff]);
            // D = A x ones + C  ->  per-row partial sums, exact f32
            acc = __builtin_amdgcn_wmma_f32_16x16x4_f32(false, a, false, bones,
                                                        (short)0, acc, false, false);
        }
        // no barrier needed: tiles are wave-private; async reuse of this
        // buffer happens 2 chunks later, after these ds_loads completed
        // (compiler inserts s_wait_dscnt before the dependent WMMAs).
    }

    // C/D layout: vgpr i, lanes 0-15 -> M=i; lanes 16-31 -> M=i+8.
    // All 16 columns identical (B == ones), so lane 0 / lane 16 suffice.
    if (lane == 0) {
        #pragma unroll
        for (int i = 0; i < 8; ++i) rowsum[w * 16 + i] = acc[i];
    } else if (lane == 16) {
        #pragma unroll
        for (int i = 0; i < 8; ++i) rowsum[w * 16 + 8 + i] = acc[i];
    }
    __syncthreads();

    if (tid < 64) {
        const int   g  = blockIdx.x * 64 + tid;            // global row, < 2048
        const int   t  = targets[g];
        const float tl = att[(size_t)g * VOCAB + t];
        const float conf = 1.0f - SMOOTH;
        const float off  = SMOOTH / (float)(VOCAB - 1);
        // -(conf*tl + off*(row_sum - tl)) = -(off*row_sum + (conf-off)*tl)
        perrow[tid] = -(off * rowsum[tid] + (conf - off) * tl);
    }
    __syncthreads();
    if (tid == 0) {
        float s = 0.f;
        for (int i = 0; i < 64; ++i) s += perrow[i];  // fixed order: deterministic
        ws[blockIdx.x] = s;
    }
}

// ---------------------------------------------------------------------------
// Kernel B: CTC alpha recursion. 16 blocks (one per batch), 288 threads
// (9 waves); thread s owns extended-label state s (L = 257). alpha lives in
// LDS; emissions are gathered from global with a one-step software pipeline
// plus a t+2 prefetch (global_prefetch_b8).
// ---------------------------------------------------------------------------
__global__ __launch_bounds__(288) void ctc_kernel(const float* __restrict__ ctc,
                                                  const int*   __restrict__ targets,
                                                  float*       __restrict__ ws) {
    __shared__ float sA[257];

    const int  b   = blockIdx.x;
    const int  s   = threadIdx.x;
    const bool act = (s < 257);
    const float* lp = ctc + (size_t)b * 512 * VOCAB;

    int  ext    = 0;       // blank for even s
    bool allow2 = false;
    if (act && (s & 1)) {
        ext = targets[b * 128 + (s >> 1)];
        if (s >= 2) {
            const int em2 = targets[b * 128 + (s >> 1) - 1];
            allow2 = (ext != 0) && (ext != em2);   // ext != BLANK && ext != ext[s-2]
        }
    }

    // t = 0 init: alpha0[0] = lp[0][blank], alpha0[1] = lp[0][tgt0], else NEG
    if (act) sA[s] = (s < 2) ? lp[ext] : NEGV;
    __syncthreads();

    float e = act ? lp[(size_t)VOCAB + ext] : 0.f;   // emission for t = 1
    for (int t = 1; t < 512; ++t) {
        float en = 0.f;
        if (act && t < 511) en = lp[(size_t)(t + 1) * VOCAB + ext];     // pipeline t+1
        if (act && t < 510) __builtin_prefetch(&lp[(size_t)(t + 2) * VOCAB + ext], 0, 0);

        float a0 = NEGV, a1 = NEGV, a2 = NEGV;
        if (act) {
            a0 = sA[s];
            if (s >= 1)  a1 = sA[s - 1];
            if (allow2)  a2 = sA[s - 2];
        }
        __syncthreads();
        if (act) sA[s] = lae(lae(a0, a1), a2) + e;
        e = en;
        __syncthreads();
    }

    if (s == 0) {
        const float ll = lae(sA[256], sA[255]);
        float lb = -ll;
        if (lb > 1e20f) lb = 0.f;        // zero_infinity
        ws[32 + b] = lb;
    }
}

// ---------------------------------------------------------------------------
// Kernel C: finalize (fixed-order sums -> deterministic scalar).
// mean(att) = att_sum/2048 ; mean(ctc/S) = ctc_sum/(16*128) = ctc_sum/2048
// ---------------------------------------------------------------------------
__global__ void fin_kernel(const float* __restrict__ ws, float* __restrict__ out) {
    if (threadIdx.x == 0 && blockIdx.x == 0) {
        float as = 0.f, cs = 0.f;
        for (int i = 0; i < 32; ++i) as += ws[i];
        for (int i = 0; i < 16; ++i) cs += ws[32 + i];
        out[0] = ALPHA_W * (as / 2048.0f) + (1.0f - ALPHA_W) * (cs / 2048.0f);
    }
}

extern "C" void kernel_launch(void* const* d_in, const int* in_sizes, int n_in,
                              void* d_out, int out_size, void* d_ws, size_t ws_size,
                              hipStream_t stream) {
    (void)in_sizes; (void)n_in; (void)out_size; (void)ws_size;
    const float* att = (const float*)d_in[0];   // (16,128,5000) f32
    const float* ctc = (const float*)d_in[1];   // (16,512,5000) f32
    const int*   tgt = (const int*)d_in[2];     // (16,128) i32
    float*       ws  = (float*)d_ws;            // [0..31] att partials, [32..47] ctc

    att_ls_kernel<<<32, 128, 0, stream>>>(att, tgt, ws);
    ctc_kernel   <<<16, 288, 0, stream>>>(ctc, tgt, ws);
    fin_kernel   <<<1,  64, 0, stream>>>(ws, (float*)d_out);
}